// RSMNet_39625368273116
// MI455X (gfx1250) — compile-verified
//
#include <hip/hip_runtime.h>
#include <math.h>

typedef __attribute__((ext_vector_type(2))) float v2f;
typedef __attribute__((ext_vector_type(8))) float v8f;
typedef __attribute__((ext_vector_type(4))) int   v4i;

#define RSM_M   1000   // columns
#define RSM_N   4      // cells per column
#define RSM_TC  4000   // total cells
#define RSM_K   25     // winning columns
#define RSM_B   128
#define RSM_T   64
#define RSM_DIN 1024
#define RSM_DOUT 1024

// ---- gfx1250 async memory->LDS staging (ASYNCcnt path), guarded so a
// ---- builtin-name mismatch falls back to plain load+ds_store cleanly.
#if defined(__AMDGCN__) && \
    __has_builtin(__builtin_amdgcn_global_load_async_to_lds_b128) && \
    __has_builtin(__builtin_amdgcn_global_load_async_to_lds_b32)
#define RSM_ASYNC_LDS 1
#else
#define RSM_ASYNC_LDS 0
#endif

#if RSM_ASYNC_LDS
typedef __attribute__((address_space(1))) v4i* gp_v4i;
typedef __attribute__((address_space(3))) v4i* lp_v4i;
typedef __attribute__((address_space(1))) int* gp_i32;
typedef __attribute__((address_space(3))) int* lp_i32;

static __device__ __forceinline__ void async_ld_b128(const float* g, float* l) {
    __builtin_amdgcn_global_load_async_to_lds_b128(
        (gp_v4i)(void*)g, (lp_v4i)(void*)l, 0, 0);
}
static __device__ __forceinline__ void async_ld_b32(const float* g, float* l) {
    __builtin_amdgcn_global_load_async_to_lds_b32(
        (gp_i32)(void*)g, (lp_i32)(void*)l, 0, 0);
}
static __device__ __forceinline__ void async_wait0() {
#if __has_builtin(__builtin_amdgcn_s_wait_asynccnt)
    __builtin_amdgcn_s_wait_asynccnt(0);
#else
    asm volatile("s_wait_asynccnt 0x0" ::: "memory");
#endif
}
#endif

// ---------------------------------------------------------------------------
// Generic fp32 GEMM on the CDNA5 matrix pipe:  C[i][j] = sum_k A[i][k]*W[j][k]
// V_WMMA_F32_16X16X4_F32, 256 threads = 8 waves, block tile 32x64.
// KC = K-chunk staged through LDS: 32 (8 WMMAs/barrier, b128 staging) when
// Kdim % 32 == 0, else 8.
// MODE 0: plain store
// MODE 1: sigma epilogue: C[i][j] = acc + extra[i][j/4]   (z_a column drive)
// MODE 2: bias  epilogue: C[i][j] = acc + extra[j]
// Grid: x = ceil(Cols/64), y = Rows/32 (Rows multiple of 32 guaranteed).
// ---------------------------------------------------------------------------
template <int MODE, int KC>
__global__ __launch_bounds__(256)
void gemm_wt(const float* __restrict__ A, int lda,
             const float* __restrict__ W, int ldw,
             float* __restrict__ C, int ldc,
             int Cols, int Kdim,
             const float* __restrict__ extra, int extra_ld)
{
    constexpr int LDSS = (KC == 32) ? 36 : 10;   // padded row stride (floats)
    __shared__ float As[32 * LDSS];
    __shared__ float Bs[64 * LDSS];

    const int tid  = threadIdx.x;
    const int wave = tid >> 5;
    const int lane = tid & 31;
    const int half = lane >> 4;           // 0: lanes 0-15, 1: lanes 16-31
    const int l16  = lane & 15;

    const int i0 = blockIdx.y * 32;
    const int j0 = blockIdx.x * 64;
    const int i_sub = (wave >> 2) * 16;   // 2 i-subtiles
    const int j_sub = (wave & 3) * 16;    // 4 j-subtiles

    // staging assignment: thread -> (row = tid/8, cols tid%8 (x1 or x4))
    const int srow = tid >> 3;
    const int scol = tid & 7;
    int jr0 = j0 + srow;      if (jr0 >= Cols) jr0 = Cols - 1;
    int jr1 = j0 + srow + 32; if (jr1 >= Cols) jr1 = Cols - 1;

    v8f acc = {};

    for (int k0 = 0; k0 < Kdim; k0 += KC) {
        if constexpr (KC == 32) {
            const int c4 = scol * 4;
            const float* ga = &A[(long)(i0 + srow) * lda + k0 + c4];
            const float* g0 = &W[(long)jr0 * ldw + k0 + c4];
            const float* g1 = &W[(long)jr1 * ldw + k0 + c4];
#if RSM_ASYNC_LDS
            async_ld_b128(ga, &As[srow * LDSS + c4]);
            async_ld_b128(g0, &Bs[srow * LDSS + c4]);
            async_ld_b128(g1, &Bs[(srow + 32) * LDSS + c4]);
#else
            *(float4*)&As[srow * LDSS + c4]        = *(const float4*)ga;
            *(float4*)&Bs[srow * LDSS + c4]        = *(const float4*)g0;
            *(float4*)&Bs[(srow + 32) * LDSS + c4] = *(const float4*)g1;
#endif
            if (k0 + KC < Kdim) {
                __builtin_prefetch(g0 + KC, 0, 3);
                __builtin_prefetch(g1 + KC, 0, 3);
            }
        } else {
            const float* ga = &A[(long)(i0 + srow) * lda + k0 + scol];
            const float* g0 = &W[(long)jr0 * ldw + k0 + scol];
            const float* g1 = &W[(long)jr1 * ldw + k0 + scol];
#if RSM_ASYNC_LDS
            async_ld_b32(ga, &As[srow * LDSS + scol]);
            async_ld_b32(g0, &Bs[srow * LDSS + scol]);
            async_ld_b32(g1, &Bs[(srow + 32) * LDSS + scol]);
#else
            As[srow * LDSS + scol]        = *ga;
            Bs[srow * LDSS + scol]        = *g0;
            Bs[(srow + 32) * LDSS + scol] = *g1;
#endif
            if (k0 + KC < Kdim) {
                __builtin_prefetch(g0 + KC, 0, 3);
                __builtin_prefetch(g1 + KC, 0, 3);
            }
        }
#if RSM_ASYNC_LDS
        async_wait0();                    // this wave's async LDS writes done
#endif
        __syncthreads();                  // all waves' staging visible

        const float* ar = &As[(i_sub + l16) * LDSS];
        const float* br = &Bs[(j_sub + l16) * LDSS];
#pragma unroll
        for (int ks = 0; ks < KC; ks += 4) {
            const int o = ks + 2 * half;  // A frag: lanes 16-31 hold K=2,3
            v2f af, bf;
            af[0] = ar[o]; af[1] = ar[o + 1];
            bf[0] = br[o]; bf[1] = br[o + 1];
            acc = __builtin_amdgcn_wmma_f32_16x16x4_f32(
                false, af, false, bf, (short)0, acc, false, false);
        }
        __syncthreads();
    }

    const int jj = j0 + j_sub + l16;
    if (jj < Cols) {
#pragma unroll
        for (int r = 0; r < 8; ++r) {     // C/D: VGPR r holds m = r + 8*half
            const int ii = i0 + i_sub + half * 8 + r;
            float v = acc[r];
            if (MODE == 1) v += extra[(long)ii * extra_ld + (jj >> 2)];
            if (MODE == 2) v += extra[jj];
            C[(long)ii * ldc + jj] = v;
        }
    }
}

// ---------------------------------------------------------------------------
// Per-step winner selection + trace update. One block per batch row.
// pi = sigma*(1-phi); lam[c] = max over 4 cells; top-25 columns of lam;
// one winning cell per column (first-max argmax); y = tanh(sigma) at winners;
// psi/phi decay by 0.5 then max with y; y_col[c] = max over cells of y.
// ---------------------------------------------------------------------------
__global__ __launch_bounds__(256)
void rsm_select_update(const float* __restrict__ sigma,
                       float* __restrict__ psi,
                       float* __restrict__ phi,
                       float* __restrict__ ycol)
{
    __shared__ float lam[RSM_M];
    __shared__ int   am[RSM_M];
    __shared__ float rv[256];
    __shared__ int   ri[256];
    __shared__ int   winC[RSM_K];
    __shared__ int   winA[RSM_K];

    const int b   = blockIdx.x;
    const int tid = threadIdx.x;

    const float* srow = sigma + (long)b * RSM_TC;
    float* psrow = psi  + (long)b * RSM_TC;
    float* phrow = phi  + (long)b * RSM_TC;
    float* yrow  = ycol + (long)b * RSM_M;

    // phase 1: per-column max/argmax of inhibited drive
    for (int c = tid; c < RSM_M; c += 256) {
        const float4 s4 = *(const float4*)(srow + 4 * c);
        const float4 p4 = *(const float4*)(phrow + 4 * c);
        const float p0 = s4.x * (1.f - p4.x);
        const float p1 = s4.y * (1.f - p4.y);
        const float p2 = s4.z * (1.f - p4.z);
        const float p3 = s4.w * (1.f - p4.w);
        float bv = p0; int ba = 0;
        if (p1 > bv) { bv = p1; ba = 1; }
        if (p2 > bv) { bv = p2; ba = 2; }
        if (p3 > bv) { bv = p3; ba = 3; }
        lam[c] = bv; am[c] = ba;
    }
    __syncthreads();

    // phase 2: iterative top-25 (value desc, index asc tie-break)
    for (int r = 0; r < RSM_K; ++r) {
        float bv = -3.4e38f; int bc = RSM_M;
        for (int c = tid; c < RSM_M; c += 256) {
            const float v = lam[c];
            if (v > bv) { bv = v; bc = c; }   // ascending c -> first max kept
        }
        rv[tid] = bv; ri[tid] = bc;
        __syncthreads();
        for (int s = 128; s > 0; s >>= 1) {
            if (tid < s) {
                const float ov = rv[tid + s]; const int oc = ri[tid + s];
                if (ov > rv[tid] || (ov == rv[tid] && oc < ri[tid])) {
                    rv[tid] = ov; ri[tid] = oc;
                }
            }
            __syncthreads();
        }
        if (tid == 0) {
            const int c = ri[0];
            winC[r] = c; winA[r] = am[c];
            lam[c] = -3.4e38f;                // exclude from later rounds
        }
        __syncthreads();
    }

    // phase 3: decay traces (EPS = GAMMA = 0.5), zero y_col row
    for (int q = tid; q < RSM_TC / 4; q += 256) {
        float4* ps4 = (float4*)psrow + q;
        float4* ph4 = (float4*)phrow + q;
        float4 a = *ps4;
        a.x *= 0.5f; a.y *= 0.5f; a.z *= 0.5f; a.w *= 0.5f;
        *ps4 = a;
        float4 g = *ph4;
        g.x *= 0.5f; g.y *= 0.5f; g.z *= 0.5f; g.w *= 0.5f;
        *ph4 = g;
    }
    for (int c = tid; c < RSM_M; c += 256) yrow[c] = 0.f;
    __syncthreads();

    // phase 4: apply winners (distinct columns -> no races)
    if (tid < RSM_K) {
        const int c = winC[tid];
        const int j = 4 * c + winA[tid];
        const float y = tanhf(srow[j]);       // mask==1 at winner
        psrow[j] = fmaxf(psrow[j], y);
        phrow[j] = fmaxf(phrow[j], y);
        yrow[c]  = fmaxf(y, 0.f);             // max over cells (others are 0)
    }
}

// ---------------------------------------------------------------------------
extern "C" void kernel_launch(void* const* d_in, const int* in_sizes, int n_in,
                              void* d_out, int out_size, void* d_ws, size_t ws_size,
                              hipStream_t stream)
{
    (void)in_sizes; (void)n_in; (void)out_size; (void)ws_size;

    const float* x   = (const float*)d_in[0]; // [T,B,D_IN]
    const float* w_a = (const float*)d_in[1]; // [M, D_IN]
    const float* w_b = (const float*)d_in[2]; // [TC, TC]
    const float* w_d = (const float*)d_in[3]; // [D_OUT, M]
    const float* b_d = (const float*)d_in[4]; // [D_OUT]
    float* out = (float*)d_out;               // [T,B,D_OUT]

    // workspace layout (floats); all offsets 64B-aligned
    float* zA   = (float*)d_ws;                                   // [T*B, M]
    float* Ycol = zA   + (size_t)RSM_T * RSM_B * RSM_M;           // [T*B, M]
    float* psi  = Ycol + (size_t)RSM_T * RSM_B * RSM_M;           // [B, TC]
    float* phi  = psi  + (size_t)RSM_B * RSM_TC;                  // [B, TC]
    float* sig  = phi  + (size_t)RSM_B * RSM_TC;                  // [B, TC]

    (void)hipMemsetAsync(psi, 0, (size_t)2 * RSM_B * RSM_TC * sizeof(float),
                         stream);

    const dim3 blk(256);

    // batched feedforward drive: zA[t*B+b][m] = x[t,b,:] . w_a[m,:]
    {
        dim3 grd((RSM_M + 63) / 64, (RSM_T * RSM_B) / 32);
        gemm_wt<0, 32><<<grd, blk, 0, stream>>>(x, RSM_DIN, w_a, RSM_DIN,
                                                zA, RSM_M, RSM_M, RSM_DIN,
                                                nullptr, 0);
    }

    // sequential recurrence
    for (int t = 0; t < RSM_T; ++t) {
        const float* zA_t = zA   + (size_t)t * RSM_B * RSM_M;
        float*       yc_t = Ycol + (size_t)t * RSM_B * RSM_M;
        dim3 grd((RSM_TC + 63) / 64, RSM_B / 32);
        // sigma = psi @ w_b^T + repeat(z_a, 4)   (MEM_GAIN = 1)
        gemm_wt<1, 32><<<grd, blk, 0, stream>>>(psi, RSM_TC, w_b, RSM_TC,
                                                sig, RSM_TC, RSM_TC, RSM_TC,
                                                zA_t, RSM_M);
        rsm_select_update<<<dim3(RSM_B), blk, 0, stream>>>(sig, psi, phi, yc_t);
    }

    // batched decode: out = Ycol @ w_d^T + b_d   (K = 1000 -> KC = 8 path)
    {
        dim3 grd((RSM_DOUT + 63) / 64, (RSM_T * RSM_B) / 32);
        gemm_wt<2, 8><<<grd, blk, 0, stream>>>(Ycol, RSM_M, w_d, RSM_M,
                                               out, RSM_DOUT, RSM_DOUT, RSM_M,
                                               b_d, 0);
    }
}